// DeformConv2D_19370302505316
// MI455X (gfx1250) — compile-verified
//
#include <hip/hip_runtime.h>
#include <hip/hip_bf16.h>

// Deformable conv for MI455X (gfx1250, wave32, WMMA).
// Pipeline:
//   K0: repack W_k -> Wmat[o][k], k = n*64 + c
//   K1: offset conv (direct VALU conv, weights staged in LDS)
//   K2: fused bilinear-gather + fp32 WMMA GEMM (M=65536, N=128, K=576)
// Uses V_WMMA_F32_16X16X4_F32 (exact fp32) since the problem is
// gather/L2-bound (~11 GFLOP, x resident in 192MB L2), not math-bound.
// Round 2: gather + W-staging loops batched (4 iters) so global loads
// issue in clauses instead of load->wait(0)->fma serialization.

typedef __attribute__((ext_vector_type(2))) float v2f;
typedef __attribute__((ext_vector_type(8))) float v8f;

#define BATCH 16
#define CIN   64
#define HW    4096      // 64*64
#define OUTC  128
#define KTOT  576       // 9*64
#define LDP   34        // LDS row stride (conflict-free for frag b64 loads)

__device__ __forceinline__ v8f wmma4(v2f a, v2f b, v8f c) {
  // D(16x16,f32) = A(16x4,f32) x B(4x16,f32) + C
  return __builtin_amdgcn_wmma_f32_16x16x4_f32(
      false, a, false, b, (short)0, c, false, false);
}

// ---------------------------------------------------------------- K0: pack W
__global__ __launch_bounds__(256) void pack_wk(const float* __restrict__ Wk,
                                               float* __restrict__ Wmat) {
  int idx = blockIdx.x * 256 + threadIdx.x;          // idx = o*576 + n*64 + c
  if (idx >= OUTC * KTOT) return;
  int o = idx / KTOT, r = idx % KTOT;
  int n = r >> 6, c = r & 63;
  Wmat[idx] = Wk[o * KTOT + c * 9 + n];              // W_k[o][c][n/3][n%3]
}

// ------------------------------------------------------- K1: offset 3x3 conv
__global__ __launch_bounds__(256) void offset_conv(
    const float* __restrict__ x, const float* __restrict__ Woff,
    const float* __restrict__ boff, float* __restrict__ Off) {
  __shared__ float wl[18 * 64 * 9];                  // 41.5 KB
  const int tid = threadIdx.x;
  for (int idx = tid; idx < 18 * 64 * 9; idx += 256) {
    int oc = idx % 18, r = idx / 18;                 // r = ic*9 + kk
    wl[idx] = Woff[oc * KTOT + r];
  }
  __syncthreads();

  const int m = blockIdx.x * 256 + tid;
  const int b = m >> 12, pix = m & 4095, i = pix >> 6, j = pix & 63;
  float acc[18];
#pragma unroll
  for (int oc = 0; oc < 18; ++oc) acc[oc] = boff[oc];

  const float* xb = x + (size_t)b * (CIN * HW);
  for (int ic = 0; ic < CIN; ++ic) {
    const float* xc = xb + ic * HW;
    const float* wrow = &wl[ic * 9 * 18];
    // load the full 3x3 neighborhood first (independent loads in flight)
    float v[9];
#pragma unroll
    for (int ki = 0; ki < 3; ++ki) {
      const int ii = i + ki - 1;
      const bool oki = (unsigned)ii < 64u;
#pragma unroll
      for (int kj = 0; kj < 3; ++kj) {
        const int jj = j + kj - 1;
        v[ki * 3 + kj] =
            (oki && (unsigned)jj < 64u) ? xc[ii * 64 + jj] : 0.f;
      }
    }
#pragma unroll
    for (int kk = 0; kk < 9; ++kk) {
      const float* w = wrow + kk * 18;               // LDS broadcast reads
#pragma unroll
      for (int oc = 0; oc < 18; ++oc) acc[oc] = fmaf(v[kk], w[oc], acc[oc]);
    }
  }
#pragma unroll
  for (int oc = 0; oc < 18; ++oc)
    Off[((size_t)(b * 18 + oc)) * HW + pix] = acc[oc];
}

// ------------------------ K2: fused bilinear-gather + WMMA fp32 GEMM
// Block: 256 thr (8 waves), tile 128(o) x 128(m); wave = 32(o) x 64(m).
__global__ __launch_bounds__(256) void deform_wmma_gemm(
    const float* __restrict__ x, const float* __restrict__ Off,
    const float* __restrict__ Wmat, const float* __restrict__ bk,
    float* __restrict__ out) {
  __shared__ float At[128 * LDP];                    // m-major, k cols
  __shared__ float Wt[128 * LDP];                    // o-major, k cols

  const int tid  = threadIdx.x;
  const int lane = tid & 31, lr = lane & 15;
  const int wv   = tid >> 5;
  const int o0   = (wv & 3) << 5;                    // wave o base (32 rows)
  const int mw   = (wv >> 2) << 6;                   // wave m base (64 cols)
  const int kb_lane = (lane >> 4) << 1;              // half-wave K offset 0/2
  const int mb   = blockIdx.x << 7;                  // block m base
  const int b    = mb >> 12;                         // whole block same batch
  const int mloc = tid & 127;                        // loader: own pixel
  const int kcol = tid >> 7;                         // loader: k parity 0/1
  const int pix  = (mb & 4095) + mloc;
  const float* xb = x + (size_t)b * (CIN * HW);

  v8f c00 = {}, c01 = {}, c02 = {}, c03 = {};
  v8f c10 = {}, c11 = {}, c12 = {}, c13 = {};

  float w00 = 0.f, w10 = 0.f, w01 = 0.f, w11 = 0.f;
  int i00 = 0, i10 = 0, i01 = 0, i11 = 0;

  for (int chunk = 0; chunk < 18; ++chunk) {
    const int kc = chunk << 5;
    __syncthreads();                                 // prior MMA done

    if ((chunk & 1) == 0) {                          // new tap n: new params
      const int n = chunk >> 1;
      const float ox = Off[((size_t)(b * 18 + n)) * HW + pix];
      const float oy = Off[((size_t)(b * 18 + 9 + n)) * HW + pix];
      float px = (float)(pix >> 6) + (float)(n / 3 - 1) + ox;
      float py = (float)(pix & 63) + (float)(n % 3 - 1) + oy;
      px = fminf(fmaxf(px, 0.f), 63.f);
      py = fminf(fmaxf(py, 0.f), 63.f);
      const float fx = floorf(px), fy = floorf(py);
      const float tx = px - fx, ty = py - fy;
      const int x0 = (int)fx, y0 = (int)fy;
      const int x1 = min(x0 + 1, 63), y1 = min(y0 + 1, 63);
      w00 = (1.f - tx) * (1.f - ty); w10 = tx * (1.f - ty);
      w01 = (1.f - tx) * ty;         w11 = tx * ty;
      i00 = x0 * 64 + y0; i10 = x1 * 64 + y0;
      i01 = x0 * 64 + y1; i11 = x1 * 64 + y1;
    }

    // stage W tile: Wt[o][k]; batch 4 iters so loads overlap
#pragma unroll
    for (int it = 0; it < 16; it += 4) {
      float wtmp[4];
#pragma unroll
      for (int u = 0; u < 4; ++u) {
        const int idx = ((it + u) << 8) + tid;
        wtmp[u] = Wmat[(idx >> 5) * KTOT + kc + (idx & 31)];
      }
#pragma unroll
      for (int u = 0; u < 4; ++u) {
        const int idx = ((it + u) << 8) + tid;
        Wt[(idx >> 5) * LDP + (idx & 31)] = wtmp[u];
      }
    }

    // stage A tile via fused bilinear gather: At[m][k], c = (kc&63)+k.
    // Batch 4 k-columns: 16 independent tap loads in flight, then FMAs.
    const int c0 = kc & 63;
#pragma unroll
    for (int it = 0; it < 16; it += 4) {
      float t0[4], t1[4], t2[4], t3[4];
#pragma unroll
      for (int u = 0; u < 4; ++u) {
        const int k = ((it + u) << 1) + kcol;        // uniform k per wave
        const float* p = xb + (size_t)(c0 + k) * HW;
        t0[u] = p[i00]; t1[u] = p[i10]; t2[u] = p[i01]; t3[u] = p[i11];
      }
#pragma unroll
      for (int u = 0; u < 4; ++u) {
        const int k = ((it + u) << 1) + kcol;
        At[mloc * LDP + k] =
            w00 * t0[u] + w10 * t1[u] + w01 * t2[u] + w11 * t3[u];
      }
    }
    __syncthreads();                                 // tiles visible

    // 8 WMMA per K-step, 8 K-steps per chunk
#pragma unroll
    for (int kk = 0; kk < 32; kk += 4) {
      const int kb = kk + kb_lane;
      const v2f a0 = *(const v2f*)&Wt[(o0 + lr) * LDP + kb];
      const v2f a1 = *(const v2f*)&Wt[(o0 + 16 + lr) * LDP + kb];
      const v2f b0 = *(const v2f*)&At[(mw + lr) * LDP + kb];
      const v2f b1 = *(const v2f*)&At[(mw + 16 + lr) * LDP + kb];
      const v2f b2 = *(const v2f*)&At[(mw + 32 + lr) * LDP + kb];
      const v2f b3 = *(const v2f*)&At[(mw + 48 + lr) * LDP + kb];
      c00 = wmma4(a0, b0, c00); c01 = wmma4(a0, b1, c01);
      c02 = wmma4(a0, b2, c02); c03 = wmma4(a0, b3, c03);
      c10 = wmma4(a1, b0, c10); c11 = wmma4(a1, b1, c11);
      c12 = wmma4(a1, b2, c12); c13 = wmma4(a1, b3, c13);
    }
  }

  // epilogue: C layout VGPR r -> o=r (+8 hi half-wave), lanes -> m
  const int hiSel = (lane >> 4) << 3;
  float* outb = out + (size_t)(b * OUTC) * HW;
#define ST_TILE(CV, OT, MT)                                          \
  {                                                                  \
    const int ob = o0 + ((OT) << 4) + hiSel;                         \
    const int pc = (mb & 4095) + mw + ((MT) << 4) + lr;              \
    _Pragma("unroll") for (int r = 0; r < 8; ++r) {                  \
      const int o = ob + r;                                          \
      outb[(size_t)o * HW + pc] = CV[r] + bk[o];                     \
    }                                                                \
  }
  ST_TILE(c00, 0, 0) ST_TILE(c01, 0, 1) ST_TILE(c02, 0, 2) ST_TILE(c03, 0, 3)
  ST_TILE(c10, 1, 0) ST_TILE(c11, 1, 1) ST_TILE(c12, 1, 2) ST_TILE(c13, 1, 3)
#undef ST_TILE
}

// ------------------------------------------------------------------ launcher
extern "C" void kernel_launch(void* const* d_in, const int* in_sizes, int n_in,
                              void* d_out, int out_size, void* d_ws,
                              size_t ws_size, hipStream_t stream) {
  const float* x    = (const float*)d_in[0];
  const float* Woff = (const float*)d_in[1];
  const float* boff = (const float*)d_in[2];
  const float* Wk   = (const float*)d_in[3];
  const float* bk   = (const float*)d_in[4];
  float* out = (float*)d_out;

  // workspace: offsets (16*18*4096 f32 = 4.5MB) + packed W (128*576 f32)
  float* Off  = (float*)d_ws;
  float* Wmat = Off + (size_t)BATCH * 18 * HW;

  pack_wk<<<(OUTC * KTOT + 255) / 256, 256, 0, stream>>>(Wk, Wmat);
  offset_conv<<<(BATCH * HW) / 256, 256, 0, stream>>>(x, Woff, boff, Off);
  deform_wmma_gemm<<<(BATCH * HW) / 128, 256, 0, stream>>>(x, Off, Wmat, bk,
                                                           out);
}